// MeanField_23854248362801
// MI455X (gfx1250) — compile-verified
//
#include <hip/hip_runtime.h>
#include <cstdint>

// ---------------- problem constants ----------------
#define BB    8
#define HH    256
#define WW    256
#define HWSZ  (HH*WW)          // 65536
#define K2    49
#define PADR  3
#define PH    (HH + 2*PADR)    // 262
#define PW    (WW + 2*PADR)    // 262
#define PHW   (PH*PW)          // 68644
#define TILE  32
#define HALO  (TILE + 2*PADR)  // 38
#define LOWC  0.1f
#define HIGHC 0.9f
#define INV2Z2 50.0f           // 1/(2*0.1^2)
#define NITER 10

struct alignas(8) F2 { float x, y; };

typedef __attribute__((ext_vector_type(4))) unsigned int u32x4;
typedef __attribute__((ext_vector_type(4))) int          i32x4;
typedef __attribute__((ext_vector_type(8))) int          i32x8;

__device__ __forceinline__ unsigned rfl(unsigned v) {
  return (unsigned)__builtin_amdgcn_readfirstlane((int)v);
}

// ------- Tensor Data Mover: 2-D tile global -> LDS (CDNA5 ISA §8 D#) -------
// group0 = {count=1, lds_addr, global_addr lo, global_addr hi | type=2}
// group1 = {data_size, tensor_dim0/1, tile_dim0/1, tensor_dim0_stride}
// dsz_code: 2 -> 4B elements, 3 -> 8B elements. dims/strides in elements.
__device__ __forceinline__ void tdm_load_tile(uint32_t lds_byte_addr,
                                              uint64_t gaddr,
                                              uint32_t tile_w, uint32_t tile_h,
                                              uint32_t tensor_w, uint32_t tensor_h,
                                              uint32_t stride_elems,
                                              uint32_t dsz_code) {
  u32x4 g0;
  g0[0] = rfl(1u);                                             // count=1 (valid), user mode
  g0[1] = rfl(lds_byte_addr);                                  // lds_addr (bytes)
  g0[2] = rfl((uint32_t)gaddr);                                // global_addr[31:0]
  g0[3] = rfl(((uint32_t)(gaddr >> 32) & 0x01FFFFFFu) | (2u << 30)); // addr[56:32] | type=2

  i32x8 g1;
  g1[0] = (int)rfl(dsz_code << 16);                            // wg_mask=0, data_size
  g1[1] = (int)rfl((tensor_w & 0xFFFFu) << 16);                // tensor_dim0[15:0]
  g1[2] = (int)rfl(((tensor_w >> 16) & 0xFFFFu) |
                   ((tensor_h & 0xFFFFu) << 16));              // dim0 hi | tensor_dim1 lo
  g1[3] = (int)rfl(((tensor_h >> 16) & 0xFFFFu) |
                   ((tile_w & 0xFFFFu) << 16));                // dim1 hi | tile_dim0
  g1[4] = (int)rfl(tile_h & 0xFFFFu);                          // tile_dim1, tile_dim2=0
  g1[5] = (int)rfl(stride_elems);                              // tensor_dim0_stride lo
  g1[6] = 0;                                                   // stride hi | dim1_stride lo
  g1[7] = 0;

  i32x4 gz = {0, 0, 0, 0};
#if __clang_major__ >= 23
  i32x8 gz8 = {0, 0, 0, 0, 0, 0, 0, 0};
  __builtin_amdgcn_tensor_load_to_lds(g0, g1, gz, gz, gz8, 0);
#else
  __builtin_amdgcn_tensor_load_to_lds(g0, g1, gz, gz, 0);
#endif
}

// shared epilogue: normalize, clip, write new log-state (+ optional threshold)
__device__ __forceinline__ void mf_epilogue(float a0, float a1,
                                            const float* __restrict__ tgt,
                                            F2* __restrict__ out_lg,
                                            float* __restrict__ d_out,
                                            int b, int y, int x, int last) {
  const int p = y * WW + x;
  const float e0 = __expf(-a0);
  const float e1 = __expf(-a1) * tgt[(size_t)b * HWSZ + p];
  float o1 = e1 / (1e-6f + e0 + e1);
  o1 = fminf(fmaxf(o1, LOWC), HIGHC);
  out_lg[(size_t)b * PHW + (size_t)(y + PADR) * PW + (x + PADR)] =
      F2{-__logf(1.f - o1), -__logf(o1)};
  if (last) d_out[(size_t)b * HWSZ + p] = (o1 > 0.5f) ? 1.f : 0.f;
}

// ---------------- kernel: pairwise Gaussian weights (fast path, once) ----------------
__global__ __launch_bounds__(256) void mf_pairwise_kernel(
    const float* __restrict__ fm, float* __restrict__ Wk) {
  const int idx = blockIdx.x * 256 + threadIdx.x;  // b*HW + p, exact grid
  const int b = idx >> 16;
  const int p = idx & (HWSZ - 1);
  const int y = p >> 8, x = p & 255;
  const float* f = fm + (size_t)b * 3 * HWSZ;
  const float c0 = f[p] + 10.f;
  const float c1 = f[HWSZ + p] + 10.f;
  const float c2 = f[2 * HWSZ + p] + 10.f;
  const size_t wb = (size_t)b * K2 * HWSZ;
#pragma unroll
  for (int i = 0; i < 7; ++i) {
    const int ny = y + i - PADR;
#pragma unroll
    for (int j = 0; j < 7; ++j) {
      const int nx = x + j - PADR;
      float n0 = 0.f, n1 = 0.f, n2 = 0.f;          // zero pad of (fm+10), matches unfold
      if (ny >= 0 && ny < HH && nx >= 0 && nx < WW) {
        const int q = ny * WW + nx;
        n0 = f[q] + 10.f; n1 = f[HWSZ + q] + 10.f; n2 = f[2 * HWSZ + q] + 10.f;
      }
      const float d0 = n0 - c0, d1 = n1 - c1, d2 = n2 - c2;
      const float s2 = d0 * d0 + d1 * d1 + d2 * d2;
      Wk[wb + (size_t)(i * 7 + j) * HWSZ + p] = __expf(-INV2Z2 * s2);
    }
  }
}

// ---------------- kernel: padded (fm+10) planes (fused path, once) ----------------
__global__ __launch_bounds__(256) void mf_padfm_kernel(
    const float* __restrict__ fm, float* __restrict__ fmp) {
  const int idx = blockIdx.x * 256 + threadIdx.x;   // over BB*3*PHW
  if (idx >= BB * 3 * PHW) return;
  const int bc = idx / PHW;                          // b*3 + c
  const int r  = idx - bc * PHW;
  const int py = r / PW;
  const int px = r - py * PW;
  float v = 0.f;                                     // zero border == unfold zero-pad
  const int y = py - PADR, x = px - PADR;
  if (y >= 0 && y < HH && x >= 0 && x < WW)
    v = fm[(size_t)bc * HWSZ + y * WW + x] + 10.f;
  fmp[idx] = v;
}

// -------- kernel: init padded log-state buffers (A = lg(s0), B = 0) --------
__global__ __launch_bounds__(256) void mf_init_kernel(
    const float* __restrict__ seg, const float* __restrict__ tgt,
    F2* __restrict__ bufA, F2* __restrict__ bufB) {
  const int idx = blockIdx.x * 256 + threadIdx.x;
  if (idx >= BB * PHW) return;
  const int b = idx / PHW;
  const int r = idx - b * PHW;
  const int py = r / PW;
  const int px = r - py * PW;
  bufB[idx] = F2{0.f, 0.f};                         // pre-zero ping-pong halo
  F2 v{0.f, 0.f};
  const int y = py - PADR, x = px - PADR;
  if (y >= 0 && y < HH && x >= 0 && x < WW) {
    const size_t q = (size_t)b * HWSZ + y * WW + x;
    float s = seg[q] * tgt[q];
    s = fminf(fmaxf(s, LOWC), HIGHC);
    v = F2{-__logf(1.f - s), -__logf(s)};
  }
  bufA[idx] = v;
}

// ---------------- kernel: one mean-field iteration (fast path) ----------------
__global__ __launch_bounds__(256) void mf_step_kernel(
    const F2* __restrict__ in_lg, F2* __restrict__ out_lg,
    const float* __restrict__ Wk, const float* __restrict__ tgt,
    float* __restrict__ d_out, int last) {
  __shared__ F2 tile[HALO * HALO];                  // 38x38 float2 = 11.55 KB

  const int b  = blockIdx.z;
  const int X0 = blockIdx.x * TILE;
  const int Y0 = blockIdx.y * TILE;

  // Async halo-tile load via TDM: each of the 8 waves DMAs its row slice.
  // TENSOR ops ignore EXEC, so every wave issues exactly one descriptor.
  {
    const uint32_t wid  = (uint32_t)(threadIdx.x >> 5);
    const uint32_t r0   = wid * 5u;
    const uint32_t rows = (wid < 7u) ? 5u : 3u;     // 7*5 + 3 = 38 rows
    const uint64_t ga = (uint64_t)(uintptr_t)(in_lg + (size_t)b * PHW +
                                              (size_t)(Y0 + (int)r0) * PW + X0);
    const uint32_t la = (uint32_t)(uintptr_t)&tile[r0 * HALO];
    tdm_load_tile(la, ga, HALO, rows, PW, PH, PW, 3u);
  }

  const int tx  = threadIdx.x & 31;
  const int ty0 = (threadIdx.x >> 5) * 4;
  const size_t wb = (size_t)b * K2 * HWSZ;

  // warm the weight-table path while the TDM runs
  __builtin_prefetch(&Wk[wb + (size_t)(Y0 + ty0) * WW + (X0 + tx)], 0, 1);

  __builtin_amdgcn_s_wait_tensorcnt(0);             // my slice landed in LDS
  __syncthreads();                                  // everyone's slice landed

  for (int rr = 0; rr < 4; ++rr) {
    const int ty = ty0 + rr;
    const int y = Y0 + ty, x = X0 + tx;
    const int p = y * WW + x;
    float a0 = 0.f, a1 = 0.f;
#pragma unroll
    for (int i = 0; i < 7; ++i) {
#pragma unroll
      for (int j = 0; j < 7; ++j) {
        const float w  = Wk[wb + (size_t)(i * 7 + j) * HWSZ + p];  // coalesced b32
        const F2    lg = tile[(ty + i) * HALO + (tx + j)];         // ds_load_b64/b128
        a0 = __builtin_fmaf(w, lg.x, a0);
        a1 = __builtin_fmaf(w, lg.y, a1);
      }
    }
    mf_epilogue(a0, a1, tgt, out_lg, d_out, b, y, x, last);
  }
}

// ------- kernel: one mean-field iteration, fused weight recompute (small-ws path) -------
__global__ __launch_bounds__(256) void mf_step_fused_kernel(
    const F2* __restrict__ in_lg, F2* __restrict__ out_lg,
    const float* __restrict__ fmp, const float* __restrict__ tgt,
    float* __restrict__ d_out, int last) {
  __shared__ F2    tlg[HALO * HALO];                // 11.55 KB
  __shared__ float tfm[3][HALO * HALO];             // 17.33 KB

  const int b  = blockIdx.z;
  const int X0 = blockIdx.x * TILE;
  const int Y0 = blockIdx.y * TILE;

  // 4 TDM descriptors per wave: log-state tile (8B elems) + 3 fm planes (4B elems)
  {
    const uint32_t wid  = (uint32_t)(threadIdx.x >> 5);
    const uint32_t r0   = wid * 5u;
    const uint32_t rows = (wid < 7u) ? 5u : 3u;
    const uint64_t ga = (uint64_t)(uintptr_t)(in_lg + (size_t)b * PHW +
                                              (size_t)(Y0 + (int)r0) * PW + X0);
    tdm_load_tile((uint32_t)(uintptr_t)&tlg[r0 * HALO], ga, HALO, rows, PW, PH, PW, 3u);
#pragma unroll
    for (int c = 0; c < 3; ++c) {
      const uint64_t gf = (uint64_t)(uintptr_t)(fmp + ((size_t)(b * 3 + c) * PHW) +
                                                (size_t)(Y0 + (int)r0) * PW + X0);
      tdm_load_tile((uint32_t)(uintptr_t)&tfm[c][r0 * HALO], gf, HALO, rows, PW, PH, PW, 2u);
    }
  }

  const int tx  = threadIdx.x & 31;
  const int ty0 = (threadIdx.x >> 5) * 4;

  __builtin_amdgcn_s_wait_tensorcnt(0);
  __syncthreads();

  for (int rr = 0; rr < 4; ++rr) {
    const int ty = ty0 + rr;
    const int y = Y0 + ty, x = X0 + tx;
    const int ci = (ty + PADR) * HALO + (tx + PADR);
    const float c0 = tfm[0][ci], c1 = tfm[1][ci], c2 = tfm[2][ci];
    float a0 = 0.f, a1 = 0.f;
#pragma unroll
    for (int i = 0; i < 7; ++i) {
#pragma unroll
      for (int j = 0; j < 7; ++j) {
        const int t = (ty + i) * HALO + (tx + j);
        const float d0 = tfm[0][t] - c0;
        const float d1 = tfm[1][t] - c1;
        const float d2 = tfm[2][t] - c2;
        const float w  = __expf(-INV2Z2 * (d0 * d0 + d1 * d1 + d2 * d2));
        const F2 lg = tlg[t];
        a0 = __builtin_fmaf(w, lg.x, a0);
        a1 = __builtin_fmaf(w, lg.y, a1);
      }
    }
    mf_epilogue(a0, a1, tgt, out_lg, d_out, b, y, x, last);
  }
}

// ---------------- host launcher ----------------
extern "C" void kernel_launch(void* const* d_in, const int* in_sizes, int n_in,
                              void* d_out, int out_size, void* d_ws, size_t ws_size,
                              hipStream_t stream) {
  const float* fm  = (const float*)d_in[0];   // (8,3,256,256)
  const float* seg = (const float*)d_in[1];   // (8,256,256)
  const float* tgt = (const float*)d_in[2];   // (8,256,256)
  float* out = (float*)d_out;
  char* ws = (char*)d_ws;

  const size_t WK_BYTES  = (size_t)BB * K2 * HWSZ * sizeof(float);   // 102.8 MB
  const size_t LG_BYTES  = (size_t)BB * PHW * sizeof(F2);            // 4.39 MB
  const size_t FMP_BYTES = (size_t)BB * 3 * PHW * sizeof(float);     // 6.59 MB
  const bool fast = (ws_size >= WK_BYTES + 2 * LG_BYTES);            // deterministic

  if (fast) {
    float* Wk  = (float*)ws;
    F2*   bufA = (F2*)(ws + WK_BYTES);
    F2*   bufB = bufA + (size_t)BB * PHW;
    hipLaunchKernelGGL(mf_pairwise_kernel, dim3((BB * HWSZ) / 256), dim3(256), 0, stream,
                       fm, Wk);
    hipLaunchKernelGGL(mf_init_kernel, dim3((BB * PHW + 255) / 256), dim3(256), 0, stream,
                       seg, tgt, bufA, bufB);
    for (int it = 0; it < NITER; ++it) {
      const F2* src = (it & 1) ? bufB : bufA;
      F2*       dst = (it & 1) ? bufA : bufB;
      hipLaunchKernelGGL(mf_step_kernel, dim3(WW / TILE, HH / TILE, BB), dim3(256), 0,
                         stream, src, dst, Wk, tgt, out, (it == NITER - 1) ? 1 : 0);
    }
  } else {
    float* fmp = (float*)ws;
    F2*   bufA = (F2*)(ws + FMP_BYTES);
    F2*   bufB = bufA + (size_t)BB * PHW;
    hipLaunchKernelGGL(mf_padfm_kernel, dim3((BB * 3 * PHW + 255) / 256), dim3(256), 0,
                       stream, fm, fmp);
    hipLaunchKernelGGL(mf_init_kernel, dim3((BB * PHW + 255) / 256), dim3(256), 0, stream,
                       seg, tgt, bufA, bufB);
    for (int it = 0; it < NITER; ++it) {
      const F2* src = (it & 1) ? bufB : bufA;
      F2*       dst = (it & 1) ? bufA : bufB;
      hipLaunchKernelGGL(mf_step_fused_kernel, dim3(WW / TILE, HH / TILE, BB), dim3(256), 0,
                         stream, src, dst, fmp, tgt, out, (it == NITER - 1) ? 1 : 0);
    }
  }
}